// DynamicUncertaintyGCN_39213051412671
// MI455X (gfx1250) — compile-verified
//
#include <hip/hip_runtime.h>
#include <math.h>

#define BB 8
#define CC 256
#define NN 4096
#define KNN 8
#define MM (BB * NN)  // 32768 total rows

typedef __attribute__((ext_vector_type(2))) float v2f;
typedef __attribute__((ext_vector_type(8))) float v8f;

__device__ __forceinline__ float gelu_exact(float x) {
  return 0.5f * x * (1.0f + erff(x * 0.70710678118654752440f));
}

// ---------------------------------------------------------------------------
// prep: fea [B][C][N] -> x [B][N][C] (tiled transpose) and fm[N][C] = mean_b
// ---------------------------------------------------------------------------
__global__ void prep_kernel(const float* __restrict__ fea,
                            float* __restrict__ x, float* __restrict__ fm) {
  __shared__ float lds[32][33];
  const int n0 = blockIdx.x * 32;
  const int c0 = blockIdx.y * 32;
  const int tx = threadIdx.x;  // n_local on load
  const int ty = threadIdx.y;  // 0..7
  float sum[4] = {0.f, 0.f, 0.f, 0.f};
  for (int b = 0; b < BB; ++b) {
    __syncthreads();
#pragma unroll
    for (int r = 0; r < 4; ++r) {
      int cl = ty + r * 8;
      float v = fea[((size_t)(b * CC + c0 + cl)) * NN + n0 + tx];
      lds[tx][cl] = v;  // [n_local][c_local]
      sum[r] += v;
    }
    __syncthreads();
#pragma unroll
    for (int r = 0; r < 4; ++r) {
      int nl = ty + r * 8;
      x[((size_t)(b * NN + n0 + nl)) * CC + c0 + tx] = lds[nl][tx];
    }
  }
  __syncthreads();
#pragma unroll
  for (int r = 0; r < 4; ++r) lds[tx][ty + r * 8] = sum[r];
  __syncthreads();
#pragma unroll
  for (int r = 0; r < 4; ++r) {
    int nl = ty + r * 8;
    fm[((size_t)(n0 + nl)) * CC + c0 + tx] = lds[nl][tx] * 0.125f;
  }
}

// ---------------------------------------------------------------------------
// sqf[n] = sum_c fm[n][c]^2
// ---------------------------------------------------------------------------
__global__ void sqf_kernel(const float* __restrict__ fm, float* __restrict__ sqf) {
  __shared__ float red[256];
  const int r = blockIdx.x;
  float v = fm[(size_t)r * CC + threadIdx.x];
  red[threadIdx.x] = v * v;
  __syncthreads();
  for (int s = 128; s > 0; s >>= 1) {
    if (threadIdx.x < s) red[threadIdx.x] += red[threadIdx.x + s];
    __syncthreads();
  }
  if (threadIdx.x == 0) sqf[r] = red[0];
}

// ---------------------------------------------------------------------------
// knn: per 16-row block, stream all 16-col tiles of fm@fm^T via f32 WMMA,
// maintain stable per-row top-8 of combined = 0.7*spatial + 0.3*feature dist
// ---------------------------------------------------------------------------
__global__ void __launch_bounds__(256)
knn_kernel(const float* __restrict__ fm, const float* __restrict__ sqf,
           int* __restrict__ gidx) {
  __shared__ float Alds[16 * 260];
  __shared__ float sqrow[16];
  __shared__ float wsc[8 * 16 * 17];  // per-wave transpose scratch
  __shared__ float mval[8 * 16 * 8];
  __shared__ int   midx[8 * 16 * 8];

  const int tid = threadIdx.x;
  const int wv = tid >> 5;
  const int lane = tid & 31;
  const int m = lane & 15;
  const bool lo = (lane < 16);
  const int koff = lo ? 0 : 2;
  const int rowbase = blockIdx.x * 16;

  for (int i = tid; i < 16 * CC; i += 256) {
    int r = i >> 8, c = i & 255;
    Alds[r * 260 + c] = fm[(size_t)(rowbase + r) * CC + c];
  }
  if (tid < 16) sqrow[tid] = sqf[rowbase + tid];
  __syncthreads();

  float bv[KNN];
  int bj[KNN];
#pragma unroll
  for (int k = 0; k < KNN; ++k) { bv[k] = 3.0e38f; bj[k] = 0x7fffffff; }

  float* sc = &wsc[wv * 16 * 17];
  const int yi = (rowbase + m) >> 6;
  const int xi = (rowbase + m) & 63;
  const float sqi = sqrow[m];

  for (int t = wv; t < 256; t += 8) {
    const int colbase = t * 16;
    v8f acc = {0.f, 0.f, 0.f, 0.f, 0.f, 0.f, 0.f, 0.f};
    const float* bp = &fm[(size_t)(colbase + m) * CC];
    for (int k0 = 0; k0 < CC; k0 += 4) {
      v2f a, b;
      a.x = Alds[m * 260 + k0 + koff];
      a.y = Alds[m * 260 + k0 + koff + 1];
      b.x = bp[k0 + koff];
      b.y = bp[k0 + koff + 1];
      acc = __builtin_amdgcn_wmma_f32_16x16x4_f32(false, a, false, b,
                                                  (short)0, acc, false, false);
    }
    // transpose D tile through per-wave LDS scratch: lane<16 -> rows 0..7
    const int mrow = lo ? 0 : 8;
#pragma unroll
    for (int v = 0; v < 8; ++v) sc[(mrow + v) * 17 + m] = acc[v];
    asm volatile("s_wait_dscnt 0" ::: "memory");
    if (lane < 16) {
#pragma unroll 1
      for (int nl = 0; nl < 16; ++nl) {
        int j = colbase + nl;
        float dotv = sc[lane * 17 + nl];
        float d2 = sqi + sqf[j] - 2.0f * dotv;
        float fd = sqrtf(fmaxf(d2, 0.0f));
        int dy = yi - (j >> 6), dx = xi - (j & 63);
        float sd = sqrtf((float)(dy * dy + dx * dx));
        float cmb = 0.7f * sd + 0.3f * fd;
        if (cmb < bv[KNN - 1]) {  // strict < keeps stable tie order (asc j)
          int p = KNN - 1;
          while (p > 0 && cmb < bv[p - 1]) {
            bv[p] = bv[p - 1]; bj[p] = bj[p - 1]; --p;
          }
          bv[p] = cmb; bj[p] = j;
        }
      }
    }
    asm volatile("s_wait_dscnt 0" ::: "memory");
  }

  if (lane < 16) {
#pragma unroll
    for (int k = 0; k < KNN; ++k) {
      mval[(wv * 16 + lane) * 8 + k] = bv[k];
      midx[(wv * 16 + lane) * 8 + k] = bj[k];
    }
  }
  __syncthreads();
  if (tid < 16) {  // stable 8-way merge (each wave list already sorted)
    int cur[8] = {0, 0, 0, 0, 0, 0, 0, 0};
    for (int k = 0; k < KNN; ++k) {
      float bestv = 3.9e38f;
      int bestj = 0x7fffffff, bw = 0;
      for (int w2 = 0; w2 < 8; ++w2) {
        int cu = cur[w2];
        if (cu < KNN) {
          float vv = mval[(w2 * 16 + tid) * 8 + cu];
          int jj = midx[(w2 * 16 + tid) * 8 + cu];
          if (vv < bestv || (vv == bestv && jj < bestj)) {
            bestv = vv; bestj = jj; bw = w2;
          }
        }
      }
      cur[bw]++;
      gidx[(rowbase + tid) * KNN + k] = bestj;
    }
  }
}

// ---------------------------------------------------------------------------
__global__ void set_kernel(float* __restrict__ p, float v, int n) {
  int i = blockIdx.x * 256 + threadIdx.x;
  if (i < n) p[i] = v;
}
__global__ void degacc_kernel(const int* __restrict__ gidx,
                              float* __restrict__ deg, int e) {
  int i = blockIdx.x * 256 + threadIdx.x;
  if (i < e) unsafeAtomicAdd(&deg[gidx[i]], 1.0f);
}
__global__ void dis_kernel(const float* __restrict__ deg,
                           float* __restrict__ dis, int n) {
  int i = blockIdx.x * 256 + threadIdx.x;
  if (i < n) dis[i] = rsqrtf(deg[i]);
}

// ---------------------------------------------------------------------------
// register-blocked fp32 WMMA GEMM: out[M x Nc] = A[M x Kd] * Wt[Kd x Nc]
// block = 128 threads (4 waves) computes a 64-row x 64-col tile.
// Each wave: 16-col strip, 4 row-tile accumulators sharing every B fragment
// (4x arithmetic intensity, 4 independent WMMA chains to hide latency).
// mode 0: plain store, mode 1: gelu(val + bias[n]). grid = (M/64, Nc/64)
// ---------------------------------------------------------------------------
__global__ void __launch_bounds__(128)
gemm_kernel(const float* __restrict__ A, const float* __restrict__ Wt,
            const float* __restrict__ bias, float* __restrict__ out,
            int Kd, int Nc, int mode) {
  __shared__ float Alds[64 * 260];
  const int tid = threadIdx.x;
  const int wv = tid >> 5;
  const int lane = tid & 31;
  const int m = lane & 15;
  const bool lo = (lane < 16);
  const int koff = lo ? 0 : 2;
  const int rowbase = blockIdx.x * 64;
  const int stride = Kd + 4;

  for (int i = tid; i < 64 * Kd; i += 128) {
    int r = i / Kd, c = i - r * Kd;
    Alds[r * stride + c] = A[(size_t)(rowbase + r) * Kd + c];
  }
  __syncthreads();

  const int ncol = blockIdx.y * 64 + wv * 16 + m;
  v8f acc[4];
#pragma unroll
  for (int rt = 0; rt < 4; ++rt)
    acc[rt] = (v8f){0.f, 0.f, 0.f, 0.f, 0.f, 0.f, 0.f, 0.f};

  for (int k0 = 0; k0 < Kd; k0 += 4) {
    v2f b;
    b.x = Wt[(size_t)(k0 + koff) * Nc + ncol];
    b.y = Wt[(size_t)(k0 + koff + 1) * Nc + ncol];
#pragma unroll
    for (int rt = 0; rt < 4; ++rt) {
      v2f a;
      a.x = Alds[(rt * 16 + m) * stride + k0 + koff];
      a.y = Alds[(rt * 16 + m) * stride + k0 + koff + 1];
      acc[rt] = __builtin_amdgcn_wmma_f32_16x16x4_f32(
          false, a, false, b, (short)0, acc[rt], false, false);
    }
  }
  const int mbase = lo ? 0 : 8;
  const float bval = (mode == 1) ? bias[ncol] : 0.0f;
#pragma unroll
  for (int rt = 0; rt < 4; ++rt) {
#pragma unroll
    for (int v = 0; v < 8; ++v) {
      float val = acc[rt][v];
      if (mode == 1) val = gelu_exact(val + bval);
      out[(size_t)(rowbase + rt * 16 + mbase + v) * Nc + ncol] = val;
    }
  }
}

// ---------------------------------------------------------------------------
// edge scatter: acc[b][dst] += dis[src]*dis[dst] * H[b][src]
// ---------------------------------------------------------------------------
__global__ void scatter_kernel(const float* __restrict__ Hb,
                               const int* __restrict__ gidx,
                               const float* __restrict__ dis,
                               float* __restrict__ accb) {
  const int e = blockIdx.x, b = blockIdx.y, c = threadIdx.x;
  const int s = e >> 3;
  const int d = gidx[e];
  const float nrm = dis[s] * dis[d];
  unsafeAtomicAdd(&accb[((size_t)b * NN + d) * CC + c],
                  nrm * Hb[((size_t)b * NN + s) * CC + c]);
}

// x += relu(acc + dis[n]^2 * H (self loop) + bias)
__global__ void gcn_epilogue_kernel(float* __restrict__ x,
                                    const float* __restrict__ accb,
                                    const float* __restrict__ Hb,
                                    const float* __restrict__ dis,
                                    const float* __restrict__ bias) {
  size_t i = (size_t)blockIdx.x * 256 + threadIdx.x;
  int c = (int)(i & (CC - 1));
  int n = (int)((i >> 8) & (NN - 1));
  float dn = dis[n];
  float v = accb[i] + dn * dn * Hb[i] + bias[c];
  x[i] += fmaxf(v, 0.0f);
}

// u[row] = sigmoid(dot(h2[row], U3) + ub3); one wave per row
__global__ void head_kernel(const float* __restrict__ h2,
                            const float* __restrict__ U3,
                            const float* __restrict__ ub3,
                            float* __restrict__ u) {
  const int row = blockIdx.x * 8 + (threadIdx.x >> 5);
  const int lane = threadIdx.x & 31;
  const float* hp = &h2[(size_t)row * 64];
  float s = hp[lane] * U3[lane] + hp[lane + 32] * U3[lane + 32];
  for (int off = 16; off > 0; off >>= 1) s += __shfl_down(s, off, 32);
  if (lane == 0) u[row] = 1.0f / (1.0f + expf(-(s + ub3[0])));
}

__global__ void final_kernel(const float* __restrict__ fea,
                             const float* __restrict__ u,
                             float* __restrict__ out) {
  size_t i = (size_t)blockIdx.x * 256 + threadIdx.x;
  int n = (int)(i & (NN - 1));
  int b = (int)(i / ((size_t)CC * NN));
  out[i] = fea[i] * (1.0f + u[b * NN + n]);
}

// ---------------------------------------------------------------------------
extern "C" void kernel_launch(void* const* d_in, const int* in_sizes, int n_in,
                              void* d_out, int out_size, void* d_ws,
                              size_t ws_size, hipStream_t stream) {
  const float* fea = (const float*)d_in[0];
  const float* W1 = (const float*)d_in[1];
  const float* b1 = (const float*)d_in[2];
  const float* W2 = (const float*)d_in[3];
  const float* b2 = (const float*)d_in[4];
  const float* W3 = (const float*)d_in[5];
  const float* b3 = (const float*)d_in[6];
  const float* U1 = (const float*)d_in[7];
  const float* ub1 = (const float*)d_in[8];
  const float* U2 = (const float*)d_in[9];
  const float* ub2 = (const float*)d_in[10];
  const float* U3 = (const float*)d_in[11];
  const float* ub3 = (const float*)d_in[12];

  char* ws = (char*)d_ws;
  size_t off = 0;
  auto alloc = [&](size_t bytes) -> void* {
    void* p = ws + off;
    off += (bytes + 255) & ~(size_t)255;
    return p;
  };
  float* fm = (float*)alloc((size_t)NN * CC * 4);
  float* sqf = (float*)alloc((size_t)NN * 4);
  int* gidx = (int*)alloc((size_t)NN * KNN * 4);
  float* deg = (float*)alloc((size_t)NN * 4);
  float* dis = (float*)alloc((size_t)NN * 4);
  float* x = (float*)alloc((size_t)MM * CC * 4);
  float* Hb = (float*)alloc((size_t)MM * CC * 4);
  float* accb = (float*)alloc((size_t)MM * CC * 4);
  float* h1 = (float*)alloc((size_t)MM * 128 * 4);
  float* h2 = (float*)alloc((size_t)MM * 64 * 4);
  float* u = (float*)alloc((size_t)MM * 4);

  prep_kernel<<<dim3(NN / 32, CC / 32), dim3(32, 8), 0, stream>>>(fea, x, fm);
  sqf_kernel<<<NN, 256, 0, stream>>>(fm, sqf);
  knn_kernel<<<NN / 16, 256, 0, stream>>>(fm, sqf, gidx);
  set_kernel<<<(NN + 255) / 256, 256, 0, stream>>>(deg, 1.0f, NN);
  degacc_kernel<<<(NN * KNN + 255) / 256, 256, 0, stream>>>(gidx, deg, NN * KNN);
  dis_kernel<<<(NN + 255) / 256, 256, 0, stream>>>(deg, dis, NN);

  const float* Wm[3] = {W1, W2, W3};
  const float* bm[3] = {b1, b2, b3};
  for (int L = 0; L < 3; ++L) {
    gemm_kernel<<<dim3(MM / 64, CC / 64), 128, 0, stream>>>(x, Wm[L], nullptr,
                                                            Hb, CC, CC, 0);
    set_kernel<<<(MM * CC + 255) / 256, 256, 0, stream>>>(accb, 0.0f, MM * CC);
    scatter_kernel<<<dim3(NN * KNN, BB), CC, 0, stream>>>(Hb, gidx, dis, accb);
    gcn_epilogue_kernel<<<MM * CC / 256, 256, 0, stream>>>(x, accb, Hb, dis,
                                                           bm[L]);
  }
  gemm_kernel<<<dim3(MM / 64, 2), 128, 0, stream>>>(x, U1, ub1, h1, CC, 128, 1);
  gemm_kernel<<<dim3(MM / 64, 1), 128, 0, stream>>>(h1, U2, ub2, h2, 128, 64, 1);
  head_kernel<<<MM / 8, 256, 0, stream>>>(h2, U3, ub3, u);
  final_kernel<<<MM * CC / 256, 256, 0, stream>>>(fea, u, (float*)d_out);
}